// MyEmbedder_84894323573251
// MI455X (gfx1250) — compile-verified
//
#include <hip/hip_runtime.h>
#include <math.h>

#define ND   300   // node embedding dim
#define RD   100   // relation embedding dim
#define PSTR 112   // P row stride: 100 proj cols + 3 w-dot cols, padded to 7*16

typedef __attribute__((ext_vector_type(2))) float v2f;
typedef __attribute__((ext_vector_type(8))) float v8f;

__device__ __forceinline__ float wred(float v) {
#pragma unroll
  for (int off = 16; off > 0; off >>= 1) v += __shfl_xor(v, off, 32);
  return v;
}
__device__ __forceinline__ float leaky(float x) { return x >= 0.f ? x : 0.2f * x; }

// ---------------- utility ----------------
__global__ void zero_f(float* p, long n) {
  long i = (long)blockIdx.x * blockDim.x + threadIdx.x;
  long st = (long)gridDim.x * blockDim.x;
  for (; i < n; i += st) p[i] = 0.f;
}

// ---------------- per-relation precompute: V = normalize(W @ r), nk, V.w2 ----
__global__ __launch_bounds__(128) void rel_prep(const float* __restrict__ rt,
    const float* __restrict__ W, const float* __restrict__ watt,
    float* __restrict__ V, float* __restrict__ nkArr, float* __restrict__ qw2) {
  __shared__ float red[128];
  int r = blockIdx.x, tid = threadIdx.x;
  const float* rr = rt + (size_t)r * RD;
  float u[3]; float n2 = 0.f;
#pragma unroll
  for (int i = 0; i < 3; ++i) {
    int d = tid + i * 128;
    float s = 0.f;
    if (d < ND) {
      const float* wr = W + (size_t)d * RD;
      for (int k = 0; k < RD; ++k) s += wr[k] * rr[k];
    }
    u[i] = s; n2 += s * s;
  }
  red[tid] = n2; __syncthreads();
  for (int s = 64; s > 0; s >>= 1) { if (tid < s) red[tid] += red[tid + s]; __syncthreads(); }
  float nk = fmaxf(sqrtf(red[0]), 1e-12f);
  float inv = 1.f / nk;
  __syncthreads();
  float q = 0.f;
#pragma unroll
  for (int i = 0; i < 3; ++i) {
    int d = tid + i * 128;
    if (d < ND) {
      float v = u[i] * inv;
      V[(size_t)r * ND + d] = v;
      q += v * watt[ND + d];
    }
  }
  red[tid] = q; __syncthreads();
  for (int s = 64; s > 0; s >>= 1) { if (tid < s) red[tid] += red[tid + s]; __syncthreads(); }
  if (tid == 0) { nkArr[r] = nk; qw2[r] = red[0]; }
}

// ---------------- G = W^T W (100x100), q1 = W^T w1 (100) ----------------
__global__ void gq_prep(const float* __restrict__ W, const float* __restrict__ watt,
                        float* __restrict__ G, float* __restrict__ q1) {
  int idx = blockIdx.x * blockDim.x + threadIdx.x;
  if (idx < RD * RD) {
    int i = idx / RD, j = idx % RD;
    float s = 0.f;
    for (int d = 0; d < ND; ++d) s += W[(size_t)d * RD + i] * W[(size_t)d * RD + j];
    G[idx] = s;
  } else if (idx < RD * RD + RD) {
    int i = idx - RD * RD;
    float s = 0.f;
    for (int d = 0; d < ND; ++d) s += W[(size_t)d * RD + i] * watt[d];
    q1[i] = s;
  }
}

// ---------------- BaugT (PSTR x ND): cols 0..99 = W_proj col, 100..102 = w1/w2/w3
__global__ void bt_prep(const float* __restrict__ W, const float* __restrict__ watt,
                        float* __restrict__ BT) {
  int idx = blockIdx.x * blockDim.x + threadIdx.x;
  if (idx >= PSTR * ND) return;
  int n = idx / ND, d = idx % ND;
  float v = 0.f;
  if (n < RD) v = W[(size_t)d * RD + n];
  else if (n < RD + 3) v = watt[(n - RD) * ND + d];
  BT[idx] = v;
}

// ---------------- WMMA f32 GEMM: P(M x 112) = A(M x 300) @ Baug(300 x 112) ----
// One wave per 16-row M tile; 7 N tiles accumulated in registers; K stepped by 4
// via V_WMMA_F32_16X16X4_F32.
__global__ __launch_bounds__(256) void gemm_p(const float* __restrict__ A,
    const float* __restrict__ BT, float* __restrict__ P, int M) {
  int wid = (blockIdx.x * blockDim.x + threadIdx.x) >> 5;
  int lane = threadIdx.x & 31;
  int m0 = wid * 16;
  if (m0 >= M) return;
  int hs = lane >> 4;          // 0 -> K pair (k,k+1); 1 -> (k+2,k+3)
  int l15 = lane & 15;
  int arowIdx = m0 + l15; if (arowIdx >= M) arowIdx = M - 1;
  const float* arow = A + (size_t)arowIdx * ND;
  v8f acc[7];
#pragma unroll
  for (int t = 0; t < 7; ++t)
#pragma unroll
    for (int i = 0; i < 8; ++i) acc[t][i] = 0.f;
  for (int k = 0; k < ND; k += 4) {
    int ka = k + hs * 2;
    v2f a; a.x = arow[ka]; a.y = arow[ka + 1];
#pragma unroll
    for (int t = 0; t < 7; ++t) {
      const float* bp = BT + (size_t)(t * 16 + l15) * ND + ka;
      v2f b; b.x = bp[0]; b.y = bp[1];
      acc[t] = __builtin_amdgcn_wmma_f32_16x16x4_f32(
          false, a, false, b, (short)0, acc[t], false, false);
    }
  }
#pragma unroll
  for (int t = 0; t < 7; ++t) {
    int col = t * 16 + l15;
#pragma unroll
    for (int i = 0; i < 8; ++i) {
      int row = m0 + i + hs * 8;
      if (row < M) P[(size_t)row * PSTR + col] = acc[t][i];
    }
  }
}

// ---------------- per-edge attention logit b + segment max (atomic key) ------
__global__ __launch_bounds__(256) void edge_b(
    const float* __restrict__ P, const float* __restrict__ rt,
    const float* __restrict__ G, const float* __restrict__ q1,
    const float* __restrict__ nkArr, const float* __restrict__ qw2,
    const int* __restrict__ parent, const int* __restrict__ child,
    const int* __restrict__ root, const int* __restrict__ rel,
    const int* __restrict__ relrp, int useRp,
    float* __restrict__ bArr, unsigned* __restrict__ bkey, int E) {
  __shared__ float sG[RD * 101];   // row-padded G to avoid LDS bank conflicts
  __shared__ float sR[8][RD];
  int tid = threadIdx.x;
  for (int i = tid; i < RD * RD; i += 256) sG[(i / RD) * 101 + (i % RD)] = G[i];
  int w = tid >> 5, lane = tid & 31;
  int e = blockIdx.x * 8 + w;
  bool act = e < E;
  int par = 0, chd = 0, rtI = 0, rl = 0;
  float t1 = 0.f, t1w = 0.f, t2 = 0.f;
  const float* Proot = P;
  const float* Ppar = P;
  if (act) {
    par = parent[e]; chd = child[e]; rtI = root[e]; rl = rel[e];
    const float* rpc = rt + (size_t)rl * RD;
    const float* rp2 = useRp ? (rt + (size_t)relrp[e] * RD) : rpc;
    Proot = P + (size_t)rtI * PSTR;
    Ppar  = P + (size_t)par * PSTR;
    for (int k = lane; k < RD; k += 32) {
      float a = rpc[k];
      float c = useRp ? a * rp2[k] : a;   // r_comp
      sR[w][k] = c;
      t1  += c * Proot[k];   // (W r_comp) . e_root
      t1w += c * q1[k];      // (W r_comp) . w1
      t2  += a * Ppar[k];    // (W r_pc) . e_parent
    }
  }
  __syncthreads();
  float np2 = 0.f;           // || W r_comp ||^2 = r^T G r
  if (act) {
    for (int k = lane; k < RD; k += 32) {
      const float* gr = sG + k * 101;
      float h = 0.f;
      for (int j = 0; j < RD; ++j) h += gr[j] * sR[w][j];
      np2 += sR[w][k] * h;
    }
  }
  t1 = wred(t1); t1w = wred(t1w); t2 = wred(t2); np2 = wred(np2);
  if (act && lane == 0) {
    float np2c = fmaxf(np2, 1e-24f);      // == max(||u||,1e-12)^2
    float nkr = nkArr[rl];
    float b = Proot[RD]     - 2.f * t1 * t1w / np2c        // e_i_proj . w1
            + Ppar[RD + 1]  - 2.f * (t2 / nkr) * qw2[rl]   // e_x_proj . w2
            + P[(size_t)chd * PSTR + RD + 2];              // e_y . w3
    b = leaky(b);
    bArr[e] = b;
    unsigned u = __float_as_uint(b);
    unsigned key = (u & 0x80000000u) ? ~u : (u | 0x80000000u); // order-preserving
    atomicMax(bkey + par, key);
  }
}

// ---------------- exp(b - max) and segment sum ----------------
__global__ void softmax_norm(const float* __restrict__ bArr, const int* __restrict__ parent,
    const unsigned* __restrict__ bkey, float* __restrict__ bexp,
    float* __restrict__ bsum, int E) {
  int e = blockIdx.x * blockDim.x + threadIdx.x;
  if (e >= E) return;
  int p = parent[e];
  unsigned k = bkey[p];
  float bm = (k & 0x80000000u) ? __uint_as_float(k ^ 0x80000000u)
                               : __uint_as_float(~k);
  float ex = expf(bArr[e] - bm);
  bexp[e] = ex;
  unsafeAtomicAdd(bsum + p, ex);
}

// ---------------- per-edge Householder update scatter-aggregate ----------------
__global__ __launch_bounds__(256) void edge_agg(
    const float* __restrict__ P, const float* __restrict__ rt,
    const float* __restrict__ V, const float* __restrict__ nkArr,
    const float* __restrict__ embs, const float* __restrict__ bexp,
    const float* __restrict__ bsum,
    const int* __restrict__ parent, const int* __restrict__ child,
    const int* __restrict__ rel, float* __restrict__ agg, int E) {
  int tid = threadIdx.x;
  int w = tid >> 5, lane = tid & 31;
  int e = blockIdx.x * 8 + w;
  if (e >= E) return;
  int par = parent[e], chd = child[e], rl = rel[e];
  const float* rpc = rt + (size_t)rl * RD;
  const float* Pch = P + (size_t)chd * PSTR;
  float s = 0.f;
  for (int k = lane; k < RD; k += 32) s += rpc[k] * Pch[k];
  s = wred(s) / nkArr[rl];                       // v_k . e_y
  float a = bexp[e] / fmaxf(bsum[par], 1e-10f);  // attention weight
  float c2 = -2.f * a * s;
  const float* ey = embs + (size_t)chd * ND;
  const float* vr = V + (size_t)rl * ND;
  float* ap = agg + (size_t)par * ND;
  for (int d = lane; d < ND; d += 32)
    unsafeAtomicAdd(ap + d, a * ey[d] + c2 * vr[d]);
}

// ---------------- out = leaky(a + b) (out may alias b) ----------------
__global__ void apply_leaky(float* __restrict__ out, const float* __restrict__ a,
                            const float* b, long n) {
  long i = (long)blockIdx.x * blockDim.x + threadIdx.x;
  long st = (long)gridDim.x * blockDim.x;
  for (; i < n; i += st) out[i] = leaky(a[i] + b[i]);
}

// ---------------- final gather ----------------
__global__ void gather_out(const float* __restrict__ src, const int* __restrict__ idx,
                           float* __restrict__ out, int B) {
  int i = blockIdx.x * blockDim.x + threadIdx.x;
  if (i >= B * ND) return;
  int r = i / ND, d = i % ND;
  out[i] = src[(size_t)idx[r] * ND + d];
}

extern "C" void kernel_launch(void* const* d_in, const int* in_sizes, int n_in,
                              void* d_out, int out_size, void* d_ws, size_t ws_size,
                              hipStream_t stream) {
  const float* node_embs = (const float*)d_in[0];
  const float* rel_table = (const float*)d_in[1];
  const float* W         = (const float*)d_in[2];
  const float* watt      = (const float*)d_in[3];
  const int* par2   = (const int*)d_in[4];
  const int* chd2   = (const int*)d_in[5];
  const int* rt2    = (const int*)d_in[6];
  const int* rel2   = (const int*)d_in[7];
  const int* relrp2 = (const int*)d_in[8];
  const int* par1   = (const int*)d_in[9];
  const int* chd1   = (const int*)d_in[10];
  const int* rt1    = (const int*)d_in[11];
  const int* rel1   = (const int*)d_in[12];
  const int* rootOut= (const int*)d_in[13];

  const int N  = in_sizes[0] / ND;   // 100000
  const int R  = in_sizes[1] / RD;   // 1000
  const int E2 = in_sizes[4];        // 200000
  const int E1 = in_sizes[9];        // 100000
  const int Bo = in_sizes[13];       // 8192

  char* wp = (char*)d_ws;
  auto take = [&](size_t bytes) {
    char* p = wp; wp += (bytes + 255) & ~(size_t)255; return p;
  };
  float* P     = (float*)take((size_t)N * PSTR * 4);   // 44.8 MB
  float* ebuf  = (float*)take((size_t)N * ND * 4);     // 120 MB
  float* agg   = (float*)take((size_t)N * ND * 4);     // 120 MB
  float* V     = (float*)take((size_t)R * ND * 4);
  float* nkArr = (float*)take((size_t)R * 4);
  float* qw2   = (float*)take((size_t)R * 4);
  float* G     = (float*)take((size_t)RD * RD * 4);
  float* q1    = (float*)take((size_t)RD * 4);
  float* BT    = (float*)take((size_t)PSTR * ND * 4);
  float* bArr  = (float*)take((size_t)E2 * 4);
  float* bexp  = (float*)take((size_t)E2 * 4);
  unsigned* bkey = (unsigned*)take((size_t)N * 4);
  float* bsum  = (float*)take((size_t)N * 4);

  // relation-level precompute (shared by both levels; depends only on inputs)
  rel_prep<<<R, 128, 0, stream>>>(rel_table, W, watt, V, nkArr, qw2);
  gq_prep<<<(RD * RD + RD + 255) / 256, 256, 0, stream>>>(W, watt, G, q1);
  bt_prep<<<(PSTR * ND + 255) / 256, 256, 0, stream>>>(W, watt, BT);

  auto run_level = [&](const float* embsIn, float* embsOut, int E,
                       const int* par, const int* chd, const int* rootI,
                       const int* rel, const int* relrp, int useRp) {
    long nTot = (long)N * ND;
    zero_f<<<2048, 256, 0, stream>>>(agg, nTot);
    zero_f<<<256, 256, 0, stream>>>(bsum, N);
    zero_f<<<256, 256, 0, stream>>>((float*)bkey, N);   // key 0 == "-inf"
    int waves = (N + 15) / 16;
    gemm_p<<<(waves + 7) / 8, 256, 0, stream>>>(embsIn, BT, P, N);
    edge_b<<<(E + 7) / 8, 256, 0, stream>>>(P, rel_table, G, q1, nkArr, qw2,
        par, chd, rootI, rel, relrp, useRp, bArr, bkey, E);
    softmax_norm<<<(E + 255) / 256, 256, 0, stream>>>(bArr, par, bkey, bexp, bsum, E);
    edge_agg<<<(E + 7) / 8, 256, 0, stream>>>(P, rel_table, V, nkArr, embsIn,
        bexp, bsum, par, chd, rel, agg, E);
    apply_leaky<<<2048, 256, 0, stream>>>(embsOut, embsIn, agg, nTot);
  };

  // level 2: r_comp = rel_table[rel_rp2] * rel_table[rel2]
  run_level(node_embs, ebuf, E2, par2, chd2, rt2, rel2, relrp2, 1);
  // level 1: r_comp == r_pc; write result in-place into agg buffer
  run_level(ebuf, agg, E1, par1, chd1, rt1, rel1, rel1, 0);

  gather_out<<<(Bo * ND + 255) / 256, 256, 0, stream>>>(agg, rootOut, (float*)d_out, Bo);
}